// DEMOWeightLayer_3083786518795
// MI455X (gfx1250) — compile-verified
//
#include <hip/hip_runtime.h>
#include <math.h>

typedef __attribute__((ext_vector_type(2))) float v2f;
typedef __attribute__((ext_vector_type(4))) float f4v;
typedef __attribute__((ext_vector_type(8))) float v8f;

// global (address_space(1)) pointer types -> force global_load/global_store
typedef const __attribute__((address_space(1))) float* gcfp;
typedef       __attribute__((address_space(1))) float* gfp;
typedef const __attribute__((address_space(1))) int*   gcip;
typedef       __attribute__((address_space(1))) int*   gip;
typedef const __attribute__((address_space(1))) f4v*   gcf4p;
typedef       __attribute__((address_space(1))) f4v*   gf4p;
typedef const __attribute__((address_space(1))) v2f*   gcv2p;

#define D128 128
#define SCAN_THREADS 1024

// ---------------------------------------------------------------------------
// Kernel 1: exclusive scan of deg -> edge offsets (dst is grouped by node).
// ---------------------------------------------------------------------------
__global__ __launch_bounds__(SCAN_THREADS)
void scan_offsets(const int* __restrict__ deg_, int* __restrict__ offs_, int n) {
    gcip deg  = (gcip)deg_;
    gip  offs = (gip)offs_;
    __shared__ int part[SCAN_THREADS];
    int t = threadIdx.x;
    int chunk = (n + SCAN_THREADS - 1) / SCAN_THREADS;
    int begin = t * chunk;
    int end = begin + chunk; if (end > n) end = n;
    int s = 0;
    for (int i = begin; i < end; ++i) s += deg[i];
    part[t] = s;
    __syncthreads();
    for (int off = 1; off < SCAN_THREADS; off <<= 1) {
        int v = (t >= off) ? part[t - off] : 0;
        __syncthreads();
        part[t] += v;
        __syncthreads();
    }
    int run = part[t] - s;              // exclusive base for this chunk
    for (int i = begin; i < end; ++i) { offs[i] = run; run += deg[i]; }
}

// ---------------------------------------------------------------------------
// Kernel 2: per-node neighbor mean (one wave32 per node, lane owns a float4
// slice of the 128-dim row) + masked self copy xm = (deg>0)*x.
// ~410 MB of gathered rows; x (25.6 MB) is L2-resident, so gathers hit L2.
// ---------------------------------------------------------------------------
__global__ __launch_bounds__(256)
void aggregate(const float* __restrict__ x_, const int* __restrict__ src_,
               const int* __restrict__ offs_, const int* __restrict__ deg_,
               float* __restrict__ nm_, float* __restrict__ xm_, int n) {
    gcfp x   = (gcfp)x_;
    gcip src = (gcip)src_;
    gcip offs = (gcip)offs_;
    gcip deg = (gcip)deg_;
    gfp  nm  = (gfp)nm_;
    gfp  xm  = (gfp)xm_;

    int node = blockIdx.x * (blockDim.x >> 5) + (threadIdx.x >> 5);
    int lane = threadIdx.x & 31;
    if (node >= n) return;

    int base = node * D128 + lane * 4;
    int dg = deg[node];
    f4v self = *(gcf4p)(x + base);

    if (dg == 0) {
        f4v z = { 0.f, 0.f, 0.f, 0.f };
        *(gf4p)(nm + base) = z;
        *(gf4p)(xm + base) = z;
        return;
    }

    int off = offs[node];
    f4v acc = { 0.f, 0.f, 0.f, 0.f };
    for (int e = 0; e < dg; ++e) {
        int s = src[off + e];
        if (e + 1 < dg) {
            int sn = src[off + e + 1];
            __builtin_prefetch(x_ + (size_t)sn * D128 + lane * 4, 0, 1);
        }
        acc += *(gcf4p)(x + (size_t)s * D128 + lane * 4);
    }
    float inv = 1.0f / (float)dg;
    acc *= inv;
    *(gf4p)(nm + base) = acc;
    *(gf4p)(xm + base) = self;
}

// ---------------------------------------------------------------------------
// Kernel 3: fused triple GEMM + bias + ELU with V_WMMA_F32_16X16X4_F32.
//   out = elu( x@Wg^T + xm@Ws^T + nm@Wl^T + b )
// One block per 16-row tile; A-slabs staged in LDS (stride 132 floats is
// conflict-free for the fp32 A-fragment pattern); 8 waves x one 16-col tile;
// 96 chained fp32 WMMAs per wave. N=50000 is divisible by 16 so EXEC is
// all-ones on every WMMA.
// ---------------------------------------------------------------------------
#define LDA 132

__global__ __launch_bounds__(256)
void gemm_fused(const float* __restrict__ x_, const float* __restrict__ xm_,
                const float* __restrict__ nm_,
                const float* __restrict__ Wg_, const float* __restrict__ Ws_,
                const float* __restrict__ Wl_,
                const float* __restrict__ bias_, float* __restrict__ out_, int n) {
    __shared__ float As[3][16 * LDA];   // 25,344 B

    int tile = blockIdx.x;
    int row0 = tile * 16;
    int t = threadIdx.x;

    // ---- scalar tail path (never taken for N=50000; runs before any WMMA
    // so the matrix path always has EXEC all-ones) ----
    if (row0 + 16 > n) {
        for (int r = row0; r < n; ++r) {
            for (int c = t; c < D128; c += 256) {
                float s = bias_[c];
                const float* xr  = x_  + (size_t)r * D128;
                const float* xmr = xm_ + (size_t)r * D128;
                const float* nmr = nm_ + (size_t)r * D128;
                const float* wg = Wg_ + (size_t)c * D128;
                const float* ws = Ws_ + (size_t)c * D128;
                const float* wl = Wl_ + (size_t)c * D128;
                for (int k = 0; k < D128; ++k)
                    s += xr[k] * wg[k] + xmr[k] * ws[k] + nmr[k] * wl[k];
                s = s > 0.f ? s : (expf(s) - 1.f);
                out_[(size_t)r * D128 + c] = s;
            }
        }
        return;
    }

    // ---- cooperative stage of the three 16x128 A-slabs into LDS ----
    {
        gcfp srcs[3] = { (gcfp)x_, (gcfp)xm_, (gcfp)nm_ };
        int li = t * 8;                // 2048 floats per slab / 256 threads
        int r = li >> 7;               // row within tile
        int c = li & 127;              // col (multiple of 8 -> same row)
        for (int b3 = 0; b3 < 3; ++b3) {
            gcfp S = srcs[b3] + (size_t)(row0 + r) * D128 + c;
            f4v v0 = *(gcf4p)(S);
            f4v v1 = *(gcf4p)(S + 4);
            float* Dst = &As[b3][r * LDA + c];
            *(f4v*)(Dst)     = v0;     // ds_store_b128
            *(f4v*)(Dst + 4) = v1;
        }
    }
    __syncthreads();

    int wave = t >> 5;                 // 0..7 -> column tile
    int lane = t & 31;
    int col0 = wave * 16;
    int mrow = lane & 15;              // A row / B column index within tile
    int koff = (lane >> 4) * 2;        // K stripe select (lanes 16-31 -> K+2)

    v8f acc = { 0.f, 0.f, 0.f, 0.f, 0.f, 0.f, 0.f, 0.f };

    gcfp Wmats[3] = { (gcfp)Wg_, (gcfp)Ws_, (gcfp)Wl_ };
    for (int b3 = 0; b3 < 3; ++b3) {
        const float* A = &As[b3][mrow * LDA];           // LDS (ds_load)
        gcfp Brow = Wmats[b3] + (size_t)(col0 + mrow) * D128;  // global
        for (int k = 0; k < D128; k += 4) {
            v2f a = *(const v2f*)(A + k + koff);        // 8B-aligned ds_load_b64
            v2f b = *(gcv2p)(Brow + k + koff);          // global_load_b64
            acc = __builtin_amdgcn_wmma_f32_16x16x4_f32(
                false, a, false, b, (short)0, acc, false, false);
        }
    }

    // ---- epilogue: bias + ELU; C layout: VGPR i -> M=i (lanes0-15) / i+8 ----
    gcfp bias = (gcfp)bias_;
    gfp  out  = (gfp)out_;
    int colg = col0 + (lane & 15);
    float bv = bias[colg];
    int rbase = row0 + (lane >> 4) * 8;
    #pragma unroll
    for (int i = 0; i < 8; ++i) {
        float v = acc[i] + bv;
        v = v > 0.f ? v : (expf(v) - 1.f);
        out[(size_t)(rbase + i) * D128 + colg] = v;
    }
}

// ---------------------------------------------------------------------------
extern "C" void kernel_launch(void* const* d_in, const int* in_sizes, int n_in,
                              void* d_out, int out_size, void* d_ws, size_t ws_size,
                              hipStream_t stream) {
    (void)n_in; (void)out_size; (void)ws_size;
    const float* x    = (const float*)d_in[0];
    const float* Wg   = (const float*)d_in[1];
    const float* Wl   = (const float*)d_in[2];
    const float* Ws   = (const float*)d_in[3];
    const float* bias = (const float*)d_in[4];
    const int*   src  = (const int*)d_in[5];
    // d_in[6] = dst: redundant (edges grouped by node; offsets from deg scan)
    const int*   deg  = (const int*)d_in[7];

    int N = in_sizes[7];

    // workspace layout: [offs: N ints][nm: N*128 f32][xm: N*128 f32]
    char* ws = (char*)d_ws;
    int* offs = (int*)ws;
    size_t off_bytes = (((size_t)N * sizeof(int)) + 255) & ~(size_t)255;
    float* nm = (float*)(ws + off_bytes);
    float* xm = nm + (size_t)N * D128;

    scan_offsets<<<1, SCAN_THREADS, 0, stream>>>(deg, offs, N);

    int wavesPerBlock = 256 / 32;
    int aggBlocks = (N + wavesPerBlock - 1) / wavesPerBlock;
    aggregate<<<aggBlocks, 256, 0, stream>>>(x, src, offs, deg, nm, xm, N);

    int tiles = (N + 15) / 16;
    gemm_fused<<<tiles, 256, 0, stream>>>(x, xm, nm, Wg, Ws, Wl, bias,
                                          (float*)d_out, N);
}